// GCN_17892833755183
// MI455X (gfx1250) — compile-verified
//
#include <hip/hip_runtime.h>
#include <hip/hip_bf16.h>

#define NNODES 50000
#define DIMIN  512
#define HIDDEN 256
#define NCLS   2
#define LDSK   516   // 512 + 4 pad -> lane l16 hits bank (4*l16)%64, conflict-free

typedef __attribute__((ext_vector_type(2))) float v2f;
typedef __attribute__((ext_vector_type(8))) float v8f;

// ---------------- degree / normalization ----------------

__global__ void k_init_deg(float* __restrict__ deg, int n) {
    int i = blockIdx.x * blockDim.x + threadIdx.x;
    if (i < n) deg[i] = 1.0f;                       // self-loop contributes 1
}

__global__ void k_deg_accum(const int* __restrict__ dst, float* __restrict__ deg, int e) {
    int i = blockIdx.x * blockDim.x + threadIdx.x;
    if (i < e) atomicAdd(&deg[dst[i]], 1.0f);
}

__global__ void k_dinv(const float* __restrict__ deg, float* __restrict__ dinv, int n) {
    int i = blockIdx.x * blockDim.x + threadIdx.x;
    if (i < n) dinv[i] = __frsqrt_rn(deg[i]);
}

// ---------------- W1 repack: Wp[k/2][col] = (W[k][col], W[k+1][col]) ----------------
// Lets each B fragment load be a single global_load_b64 instead of 2x b32.

__global__ void k_packW(const float* __restrict__ W, float2* __restrict__ Wp) {
    int idx = blockIdx.x * blockDim.x + threadIdx.x;     // over (DIMIN/2)*HIDDEN
    if (idx >= (DIMIN / 2) * HIDDEN) return;
    int kp  = idx / HIDDEN;
    int col = idx - kp * HIDDEN;
    float2 p;
    p.x = W[(size_t)(2 * kp)     * HIDDEN + col];
    p.y = W[(size_t)(2 * kp + 1) * HIDDEN + col];
    Wp[idx] = p;
}

// ---------------- GEMM1: H1 = X @ W1 via V_WMMA_F32_16X16X4_F32 ----------------
// block = 256 threads = 8 waves; block computes 16 rows x 256 cols.
// A slab (16 x 512 f32) staged in LDS (padded stride); each wave does two
// 16x16 column tiles, 128 WMMA ops each (K stepped by 4). Epilogue also
// writes AGG1 = dinv^2 * H1 (the self-loop term of the aggregation).

__global__ void __launch_bounds__(256)
k_gemm1(const float* __restrict__ X, const float2* __restrict__ Wp,
        const float* __restrict__ dinv, float* __restrict__ H, float* __restrict__ AGG) {
    __shared__ float As[16 * LDSK];

    const int row0 = blockIdx.x * 16;

    // cooperative load: 16*512 floats = 2048 float4, 8 per thread
    for (int idx = threadIdx.x; idx < (16 * DIMIN) / 4; idx += 256) {
        int r = idx >> 7;              // 128 float4 per row
        int c = (idx & 127) * 4;
        const float4 v = *(const float4*)(X + (size_t)(row0 + r) * DIMIN + c);
        *(float4*)(&As[r * LDSK + c]) = v;
    }
    __syncthreads();

    const int wave = threadIdx.x >> 5;
    const int lane = threadIdx.x & 31;
    const int l16  = lane & 15;
    const int hi   = lane >> 4;        // 0: K=k,k+1  1: K=k+2,k+3 (A); rows 0-7 vs 8-15 (D)

    for (int ct = wave; ct < HIDDEN / 16; ct += 8) {
        const int col0 = ct * 16;
        v8f c = {};
        #pragma unroll 8
        for (int k = 0; k < DIMIN; k += 4) {
            const int kk = k + 2 * hi;
            // A fragment: 16x4, lane l16 row, two consecutive K values (8B-aligned LDS read)
            const float2 af = *(const float2*)(&As[l16 * LDSK + kk]);
            v2f a; a.x = af.x; a.y = af.y;
            // B fragment: 4x16, one b64 from the K-pair-packed weight buffer
            const float2 bf = Wp[(size_t)(kk >> 1) * HIDDEN + col0 + l16];
            v2f b; b.x = bf.x; b.y = bf.y;
            c = __builtin_amdgcn_wmma_f32_16x16x4_f32(false, a, false, b,
                                                      (short)0, c, false, false);
        }
        // D layout: VGPR v -> row (hi*8 + v), col l16
        #pragma unroll
        for (int v = 0; v < 8; ++v) {
            const int row = row0 + hi * 8 + v;
            const float dv = dinv[row];
            const size_t o = (size_t)row * HIDDEN + col0 + l16;
            H[o]   = c[v];
            AGG[o] = dv * dv * c[v];
        }
    }
}

// ---------------- aggregation layer 1 ----------------

// one wave per edge: gather h1[src] (float4), scatter-add into agg1[dst]
__global__ void k_edge_agg1(const int* __restrict__ src, const int* __restrict__ dst,
                            const float* __restrict__ dinv, const float* __restrict__ H1,
                            float* __restrict__ AGG, int e) {
    int gid  = blockIdx.x * blockDim.x + threadIdx.x;
    int wid  = gid >> 5;
    int lane = gid & 31;
    if (wid >= e) return;
    int s = src[wid], d = dst[wid];
    float norm = dinv[s] * dinv[d];
    const float4* hs = (const float4*)(H1 + (size_t)s * HIDDEN);
    float* od = AGG + (size_t)d * HIDDEN;
    #pragma unroll
    for (int it = 0; it < 2; ++it) {
        int c4 = it * 32 + lane;                           // 0..63
        float4 h = hs[c4];
        int c = c4 * 4;
        atomicAdd(&od[c + 0], norm * h.x);
        atomicAdd(&od[c + 1], norm * h.y);
        atomicAdd(&od[c + 2], norm * h.z);
        atomicAdd(&od[c + 3], norm * h.w);
    }
}

// z = relu(agg1 + b1), in place
__global__ void k_bias_relu(float* __restrict__ A, const float* __restrict__ b1, int total) {
    int i = blockIdx.x * blockDim.x + threadIdx.x;
    if (i >= total) return;
    float v = A[i] + b1[i & (HIDDEN - 1)];
    A[i] = v > 0.0f ? v : 0.0f;
}

// ---------------- GEMM2: H2 = Z @ W2 (256 -> 2), scalar ----------------
// Epilogue also initializes out = dinv^2 * H2 + b2 (self-loop term, layer 2).

__global__ void k_gemm2(const float* __restrict__ Z, const float* __restrict__ W2,
                        const float* __restrict__ dinv, const float* __restrict__ b2,
                        float* __restrict__ H2, float* __restrict__ out, int n) {
    int i = blockIdx.x * blockDim.x + threadIdx.x;
    if (i >= n) return;
    const float4* z4 = (const float4*)(Z + (size_t)i * HIDDEN);
    float a0 = 0.0f, a1 = 0.0f;
    #pragma unroll 4
    for (int k4 = 0; k4 < HIDDEN / 4; ++k4) {
        float4 z = z4[k4];
        int k = k4 * 4;
        a0 += z.x * W2[2*k+0] + z.y * W2[2*k+2] + z.z * W2[2*k+4] + z.w * W2[2*k+6];
        a1 += z.x * W2[2*k+1] + z.y * W2[2*k+3] + z.z * W2[2*k+5] + z.w * W2[2*k+7];
    }
    H2[2*i]     = a0;
    H2[2*i + 1] = a1;
    float s = dinv[i]; s *= s;
    out[2*i]     = s * a0 + b2[0];
    out[2*i + 1] = s * a1 + b2[1];
}

// ---------------- aggregation layer 2 (into d_out) ----------------

__global__ void k_edge_agg2(const int* __restrict__ src, const int* __restrict__ dst,
                            const float* __restrict__ dinv, const float* __restrict__ H2,
                            float* __restrict__ out, int e) {
    int i = blockIdx.x * blockDim.x + threadIdx.x;
    if (i >= e) return;
    int s = src[i], d = dst[i];
    float norm = dinv[s] * dinv[d];
    atomicAdd(&out[2*d],     norm * H2[2*s]);
    atomicAdd(&out[2*d + 1], norm * H2[2*s + 1]);
}

// ---------------- host launcher ----------------

extern "C" void kernel_launch(void* const* d_in, const int* in_sizes, int n_in,
                              void* d_out, int out_size, void* d_ws, size_t ws_size,
                              hipStream_t stream) {
    const float* X   = (const float*)d_in[0];   // [N, 512]
    const float* W1  = (const float*)d_in[1];   // [512, 256]
    const float* b1  = (const float*)d_in[2];   // [256]
    const float* W2  = (const float*)d_in[3];   // [256, 2]
    const float* b2  = (const float*)d_in[4];   // [2]
    const int*   ei  = (const int*)d_in[5];     // [2, E] flattened: src then dst
    float*       out = (float*)d_out;           // [N, 2]

    const int N = NNODES;
    const int E = in_sizes[5] / 2;
    const int* src = ei;
    const int* dst = ei + E;

    // workspace layout (floats)
    float*  deg  = (float*)d_ws;                // N
    float*  dinv = deg  + N;                    // N
    float*  H1   = dinv + N;                    // N*256
    float*  AGG1 = H1   + (size_t)N * HIDDEN;   // N*256
    float*  H2   = AGG1 + (size_t)N * HIDDEN;   // N*2
    float2* Wp   = (float2*)(H2 + (size_t)N * NCLS);  // (512/2)*256 float2

    const int T = 256;

    // normalization
    k_init_deg <<<(N + T - 1) / T, T, 0, stream>>>(deg, N);
    k_deg_accum<<<(E + T - 1) / T, T, 0, stream>>>(dst, deg, E);
    k_dinv     <<<(N + T - 1) / T, T, 0, stream>>>(deg, dinv, N);

    // weight repack for b64 B-fragment loads
    {
        int total = (DIMIN / 2) * HIDDEN;
        k_packW<<<(total + T - 1) / T, T, 0, stream>>>(W1, Wp);
    }

    // layer 1 (GEMM epilogue also writes self-loop term into AGG1)
    k_gemm1<<<N / 16, 256, 0, stream>>>(X, Wp, dinv, H1, AGG1);
    {
        long long threads = (long long)E * 32;
        k_edge_agg1<<<(int)((threads + T - 1) / T), T, 0, stream>>>(src, dst, dinv, H1, AGG1, E);
    }
    {
        int total = N * HIDDEN;
        k_bias_relu<<<(total + T - 1) / T, T, 0, stream>>>(AGG1, b1, total);
    }

    // layer 2 (GEMM epilogue also writes self-loop term + bias into out)
    k_gemm2    <<<(N + T - 1) / T, T, 0, stream>>>(AGG1, W2, dinv, b2, H2, out, N);
    k_edge_agg2<<<(E + T - 1) / T, T, 0, stream>>>(src, dst, dinv, H2, out, E);
}